// SubjectLayers_53515292508233
// MI455X (gfx1250) — compile-verified
//
#include <hip/hip_runtime.h>

#define IN_CH       1024
#define OUT_CH      1024
#define N_SUBJECTS  32
#define B_SAMPLES   512
#define MAX_ROW_TILES ((B_SAMPLES + 15) / 16)   // 32 (worst case: all samples one subject)
#define COLS_PER_WAVE 32                        // 2 WMMA column tiles per wave
#define PF_ROWS     32                          // prefetch distance in K rows (128 KB of stream)

typedef __attribute__((ext_vector_type(2))) float v2f;
typedef __attribute__((ext_vector_type(8))) float v8f;

// ---------------------------------------------------------------------------
// Phase 1a: prime the 128 MB weight tensor into the 192 MB L2 while the
// grouping kernel runs. global_prefetch_b8 consumes no LOADcnt, so this
// kernel retires immediately after issuing; on warm replays it is ~free.
// ---------------------------------------------------------------------------
__global__ void weights_prefetch_kernel(const float* __restrict__ weights) {
    // 512 blocks x 256 threads = 131072 lanes; each lane walks 1 KB in
    // 256 B steps => covers 32*1024*1024 floats (128 MB).
    const size_t tid   = (size_t)blockIdx.x * blockDim.x + threadIdx.x;
    const char*  base  = (const char*)weights + tid * 1024;
#pragma unroll
    for (int i = 0; i < 4; ++i)
        __builtin_prefetch(base + i * 256, 0, 3);
}

// ---------------------------------------------------------------------------
// Phase 1b: bucket sample ids by subject. counts[s] = #samples, slots[s][i]=b.
// Order within a bucket is nondeterministic but each output row is computed
// independently, so the final output is deterministic.
// ---------------------------------------------------------------------------
__global__ void subject_group_kernel(const int* __restrict__ subject_idx,
                                     int* __restrict__ counts,
                                     int* __restrict__ slots) {
    int b = blockIdx.x * blockDim.x + threadIdx.x;
    if (b < B_SAMPLES) {
        int s = subject_idx[b] & (N_SUBJECTS - 1);
        int pos = atomicAdd(&counts[s], 1);
        slots[s * B_SAMPLES + pos] = b;
    }
}

// ---------------------------------------------------------------------------
// Phase 2: per-subject GEMM with V_WMMA_F32_16X16X4_F32.
// One wave (32 threads) per block; block computes 16(M) x 32(N) of the output
// for one (subject, row-tile): 2 accumulators of 16x16, K looped 1024/4.
// Low VGPR count + 2x grid.y => ~1024 concurrent waves for latency coverage.
//
// Fragment layouts (CDNA5 ISA 7.12.2, 32-bit operands):
//   A 16x4 : lanes 0-15  -> M=lane,    VGPR0=K0, VGPR1=K1
//            lanes 16-31 -> M=lane-16, VGPR0=K2, VGPR1=K3
//   B 4x16 : lanes 0-15  -> N=lane,    VGPR0=K0, VGPR1=K1
//            lanes 16-31 -> N=lane-16, VGPR0=K2, VGPR1=K3
//   C/D    : lane%16 selects N; VGPR v holds M = v + 8*(lane>=16)
// ---------------------------------------------------------------------------
__global__ void __launch_bounds__(32)
subject_gemm_wmma_kernel(const float* __restrict__ x,
                         const float* __restrict__ weights,
                         const float* __restrict__ bias,
                         const int*   __restrict__ counts,
                         const int*   __restrict__ slots,
                         float*       __restrict__ out) {
    const int s     = blockIdx.x & (N_SUBJECTS - 1);
    const int rtile = blockIdx.x >> 5;                 // row tile within subject
    const int cnt   = counts[s];
    if (rtile * 16 >= cnt) return;                     // whole wave exits

    const int lane = threadIdx.x;                      // wave32
    const int half = lane >> 4;                        // 0: K0/K1, 1: K2/K3
    const int l16  = lane & 15;

    // x row feeding this lane's A fragment (M = l16). Out-of-range rows
    // replicate the tile's first (valid) row; masked at store time.
    const int  mrow   = rtile * 16 + l16;
    const int  srcRow = slots[s * B_SAMPLES + (mrow < cnt ? mrow : rtile * 16)];
    const float* xrow = x + (size_t)srcRow * IN_CH;

    const float* W     = weights + (size_t)s * IN_CH * OUT_CH;
    const int    cbase = blockIdx.y * COLS_PER_WAVE;   // 2 column tiles of 16

    v8f acc0 = {}, acc1 = {};

#pragma unroll 4
    for (int k = 0; k < IN_CH; k += 4) {
        const int ka = k + half * 2;

        // A fragment: A[M=l16][ka], A[M=l16][ka+1]  (one b64 load, L2-resident)
        v2f a = *(const v2f*)(xrow + ka);

        // B fragments: B[ka][n], B[ka+1][n] for 2 column tiles. Default RT
        // temporal hint: 132 MB working set is resident in the 192 MB L2.
        const float* w0 = W + (size_t)ka * OUT_CH + cbase + l16;
        const float* w1 = w0 + OUT_CH;
        v2f b0 = { w0[0],  w1[0]  };
        v2f b1 = { w0[16], w1[16] };

        // Stream-ahead prefetch (global_prefetch_b8): +PF_ROWS K rows. Each
        // 128 B line at (cbase + l16) covers this wave's 32-column footprint.
        __builtin_prefetch(w0 + (size_t)PF_ROWS * OUT_CH, 0, 3);
        __builtin_prefetch(w1 + (size_t)PF_ROWS * OUT_CH, 0, 3);

        // 8 args: (neg_a, A, neg_b, B, c_mod, C, reuse_a, reuse_b)
        acc0 = __builtin_amdgcn_wmma_f32_16x16x4_f32(false, a, false, b0, (short)0, acc0, false, false);
        acc1 = __builtin_amdgcn_wmma_f32_16x16x4_f32(false, a, false, b1, (short)0, acc1, false, false);
    }

    // Bias add + masked scatter store. Lane's N is fixed per tile; VGPR v
    // holds row M = v + 8*half.
    v8f accs[2] = { acc0, acc1 };
#pragma unroll
    for (int ct = 0; ct < 2; ++ct) {
        const int   n  = cbase + ct * 16 + l16;
        const float bv = bias[s * OUT_CH + n];
        v8f acc = accs[ct];
#pragma unroll
        for (int v = 0; v < 8; ++v) {
            const int m  = v + half * 8;
            const int gm = rtile * 16 + m;
            if (gm < cnt) {
                const int r = slots[s * B_SAMPLES + gm];
                out[(size_t)r * OUT_CH + n] = acc[v] + bv;
            }
        }
    }
}

// ---------------------------------------------------------------------------
// Host launcher (graph-capture safe: memsetAsync + 3 kernels on `stream`)
// ---------------------------------------------------------------------------
extern "C" void kernel_launch(void* const* d_in, const int* in_sizes, int n_in,
                              void* d_out, int out_size, void* d_ws, size_t ws_size,
                              hipStream_t stream) {
    const float* x           = (const float*)d_in[0];
    const int*   subject_idx = (const int*)  d_in[1];
    const float* weights     = (const float*)d_in[2];
    const float* bias        = (const float*)d_in[3];
    float*       out         = (float*)d_out;

    int* counts = (int*)d_ws;                       // 32 ints
    int* slots  = counts + N_SUBJECTS;              // 32*512 ints

    hipMemsetAsync(counts, 0, N_SUBJECTS * sizeof(int), stream);

    // L2 warm-up for the weight stream (overlaps with grouping).
    weights_prefetch_kernel<<<512, 256, 0, stream>>>(weights);

    subject_group_kernel<<<(B_SAMPLES + 127) / 128, 128, 0, stream>>>(
        subject_idx, counts, slots);

    dim3 grid(N_SUBJECTS * MAX_ROW_TILES, OUT_CH / COLS_PER_WAVE);   // (1024, 32)
    subject_gemm_wmma_kernel<<<grid, 32, 0, stream>>>(
        x, weights, bias, counts, slots, out);
}